// ImplicitGraph_24919400251501
// MI455X (gfx1250) — compile-verified
//
#include <hip/hip_runtime.h>
#include <hip/hip_bf16.h>

typedef float v2f __attribute__((ext_vector_type(2)));
typedef float v4f __attribute__((ext_vector_type(4)));
typedef float v8f __attribute__((ext_vector_type(8)));

#define MDIM   64
#define NDIM   50000
#define PDIM   128
#define EDGES  400000
#define KAPPA  0.99f
#define NM     (MDIM * NDIM)          // 3,200,000 floats per [M,N] buffer
#define JT     (NDIM / 16)            // 3125 j-tiles (exact)
#define ITERS  30

// ---------------------------------------------------------------------------
// fp32-exact WMMA: D(16x16,f32) += A(16x4,f32) x B(4x16,f32)
// ---------------------------------------------------------------------------
static __device__ __forceinline__ v8f wmma4(v2f a, v2f b, v8f c) {
    return __builtin_amdgcn_wmma_f32_16x16x4_f32(
        /*neg_a=*/false, a, /*neg_b=*/false, b,
        /*c_mod=*/(short)0, c, /*reuse_a=*/false, /*reuse_b=*/false);
}

// ---------------------------------------------------------------------------
// 1) Row-wise L1-ball projection of W (64x64) onto radius KAPPA.
//    One thread per row; tiny one-shot cost.
// ---------------------------------------------------------------------------
__global__ __launch_bounds__(64) void project_linf_kernel(
        const float* __restrict__ W, float* __restrict__ Wp) {
    const int row = threadIdx.x;            // 64 threads, one row each
    float w[64], a[64], s[64];
    float sum = 0.f;
    for (int k = 0; k < 64; ++k) {
        w[k] = W[row * 64 + k];
        a[k] = fabsf(w[k]);
        s[k] = a[k];
        sum += a[k];
    }
    // insertion sort, descending
    for (int i = 1; i < 64; ++i) {
        float key = s[i];
        int k = i - 1;
        while (k >= 0 && s[k] < key) { s[k + 1] = s[k]; --k; }
        s[k + 1] = key;
    }
    float css = 0.f, cssAtRho = 0.f;
    int rho = 1;
    for (int j = 1; j <= 64; ++j) {
        css += s[j - 1];
        if (s[j - 1] * (float)j > css - KAPPA) { rho = j; cssAtRho = css; }
    }
    const float theta = (cssAtRho - KAPPA) / (float)rho;
    const bool shrink = (sum > KAPPA);
    for (int k = 0; k < 64; ++k) {
        float m = fmaxf(a[k] - theta, 0.f);
        float proj = (w[k] > 0.f) ? m : ((w[k] < 0.f) ? -m : 0.f);
        Wp[row * 64 + k] = shrink ? proj : w[k];
    }
}

// ---------------------------------------------------------------------------
// 2) B0t = (Omega @ U)^T  stored as [N,64]:  B0t[j,m] = sum_k U[k,j]*Om[m,k]
//    One wave per 16-row j-tile, 4 WMMA accumulators cover m=0..63, K=128.
// ---------------------------------------------------------------------------
__global__ __launch_bounds__(256) void gemm_omega_u_kernel(
        const float* __restrict__ U, const float* __restrict__ Om,
        float* __restrict__ B0t) {
    const int lane = threadIdx.x & 31;
    const int tile = blockIdx.x * 8 + (threadIdx.x >> 5);
    if (tile >= JT) return;                  // wave-uniform exit
    const int j0  = tile * 16;
    const int row = lane & 15;               // A row / B col / D col index
    const int kb  = (lane >> 4) * 2;         // K sub-offset: 0 or 2

    v8f acc0 = {}, acc1 = {}, acc2 = {}, acc3 = {};
    for (int k0 = 0; k0 < PDIM; k0 += 4) {
        const int k = k0 + kb;
        v2f a;
        a.x = U[(k)     * NDIM + j0 + row];
        a.y = U[(k + 1) * NDIM + j0 + row];
        v2f b0 = *(const v2f*)(Om + (0  + row) * PDIM + k);
        v2f b1 = *(const v2f*)(Om + (16 + row) * PDIM + k);
        v2f b2 = *(const v2f*)(Om + (32 + row) * PDIM + k);
        v2f b3 = *(const v2f*)(Om + (48 + row) * PDIM + k);
        acc0 = wmma4(a, b0, acc0);
        acc1 = wmma4(a, b1, acc1);
        acc2 = wmma4(a, b2, acc2);
        acc3 = wmma4(a, b3, acc3);
    }
    const int mm   = lane & 15;
    const int jjHi = (lane >> 4) * 8;
    for (int r = 0; r < 8; ++r) {
        const int j = j0 + jjHi + r;
        B0t[j * 64 + 0  + mm] = acc0[r];
        B0t[j * 64 + 16 + mm] = acc1[r];
        B0t[j * 64 + 32 + mm] = acc2[r];
        B0t[j * 64 + 48 + mm] = acc3[r];
    }
}

// ---------------------------------------------------------------------------
// 3) Yt = (Wp @ X)^T  with X kept transposed [N,64]:
//    Yt[j,m] = sum_k Xt[j,k] * Wp[m,k]   (K = 64)
// ---------------------------------------------------------------------------
__global__ __launch_bounds__(256) void gemm_w_x_kernel(
        const float* __restrict__ Xt, const float* __restrict__ Wp,
        float* __restrict__ Yt) {
    const int lane = threadIdx.x & 31;
    const int tile = blockIdx.x * 8 + (threadIdx.x >> 5);
    if (tile >= JT) return;                  // wave-uniform exit
    const int j0  = tile * 16;
    const int row = lane & 15;
    const int kb  = (lane >> 4) * 2;

    const float* Arow = Xt + (j0 + row) * 64 + kb;
    v8f acc0 = {}, acc1 = {}, acc2 = {}, acc3 = {};
    for (int k0 = 0; k0 < 64; k0 += 4) {
        v2f a  = *(const v2f*)(Arow + k0);
        v2f b0 = *(const v2f*)(Wp + (0  + row) * 64 + k0 + kb);
        v2f b1 = *(const v2f*)(Wp + (16 + row) * 64 + k0 + kb);
        v2f b2 = *(const v2f*)(Wp + (32 + row) * 64 + k0 + kb);
        v2f b3 = *(const v2f*)(Wp + (48 + row) * 64 + k0 + kb);
        acc0 = wmma4(a, b0, acc0);
        acc1 = wmma4(a, b1, acc1);
        acc2 = wmma4(a, b2, acc2);
        acc3 = wmma4(a, b3, acc3);
    }
    const int mm   = lane & 15;
    const int jjHi = (lane >> 4) * 8;
    for (int r = 0; r < 8; ++r) {
        const int j = j0 + jjHi + r;
        Yt[j * 64 + 0  + mm] = acc0[r];
        Yt[j * 64 + 16 + mm] = acc1[r];
        Yt[j * 64 + 32 + mm] = acc2[r];
        Yt[j * 64 + 48 + mm] = acc3[r];
    }
}

// ---------------------------------------------------------------------------
// 4) Sparse X@A: one wave per edge; lane handles a contiguous float2 of the
//    64-float column. Gather Yt[:,src]*val, atomic-add into Acc[:,dst].
// ---------------------------------------------------------------------------
__global__ __launch_bounds__(256) void scatter_kernel(
        const float* __restrict__ Yt,
        const int* __restrict__ src, const int* __restrict__ dst,
        const float* __restrict__ val, float* __restrict__ Acc) {
    const int lane = threadIdx.x & 31;
    const int e = blockIdx.x * 8 + (threadIdx.x >> 5);
    if (e >= EDGES) return;
    const int   s = src[e];
    const int   d = dst[e];
    const float v = val[e];
    v2f y = *(const v2f*)(Yt + s * 64 + lane * 2);
    float* p = Acc + d * 64 + lane * 2;
    atomicAdd(p,     y.x * v);
    atomicAdd(p + 1, y.y * v);
}

// ---------------------------------------------------------------------------
// 5) Elementwise helpers (float4 vectorized; NM is divisible by 4)
// ---------------------------------------------------------------------------
__global__ __launch_bounds__(256) void zero_kernel(v4f* __restrict__ p, int n4) {
    int i = blockIdx.x * 256 + threadIdx.x;
    if (i < n4) { v4f z = {}; p[i] = z; }
}
__global__ __launch_bounds__(256) void copy_kernel(
        const v4f* __restrict__ a, v4f* __restrict__ b, int n4) {
    int i = blockIdx.x * 256 + threadIdx.x;
    if (i < n4) b[i] = a[i];
}
__global__ __launch_bounds__(256) void relu_kernel(
        const v4f* __restrict__ a, v4f* __restrict__ x, int n4) {
    int i = blockIdx.x * 256 + threadIdx.x;
    if (i < n4) {
        v4f t = a[i];
        t.x = fmaxf(t.x, 0.f); t.y = fmaxf(t.y, 0.f);
        t.z = fmaxf(t.z, 0.f); t.w = fmaxf(t.w, 0.f);
        x[i] = t;
    }
}

// ---------------------------------------------------------------------------
// 6) Final transpose Xt [N,64] -> out [64,N] via padded LDS tile (64x64)
// ---------------------------------------------------------------------------
__global__ __launch_bounds__(256) void transpose_out_kernel(
        const float* __restrict__ Xt, float* __restrict__ out) {
    __shared__ float t[64][65];
    const int j0 = blockIdx.x * 64;
    for (int idx = threadIdx.x; idx < 4096; idx += 256) {
        const int j = idx >> 6;        // 0..63
        const int m = idx & 63;        // contiguous across threads -> coalesced
        t[m][j] = (j0 + j < NDIM) ? Xt[(j0 + j) * 64 + m] : 0.f;
    }
    __syncthreads();
    for (int idx = threadIdx.x; idx < 4096; idx += 256) {
        const int m  = idx >> 6;
        const int jj = idx & 63;       // contiguous -> coalesced stores
        if (j0 + jj < NDIM) out[m * NDIM + j0 + jj] = t[m][jj];
    }
}

// ---------------------------------------------------------------------------
extern "C" void kernel_launch(void* const* d_in, const int* in_sizes, int n_in,
                              void* d_out, int out_size, void* d_ws, size_t ws_size,
                              hipStream_t stream) {
    const float* W   = (const float*)d_in[0];   // [64,64]
    const float* Om  = (const float*)d_in[1];   // [64,128]
    const float* U   = (const float*)d_in[2];   // [128,50000]
    const int*   src = (const int*)d_in[3];     // [E]
    const int*   dst = (const int*)d_in[4];     // [E]
    const float* val = (const float*)d_in[5];   // [E]
    float* out = (float*)d_out;                 // [64,50000]

    // Workspace layout (floats): Wp | Yt(/B0t) | bT | Acc | Xt
    float* Wp  = (float*)d_ws;
    float* Yt  = Wp  + 64 * 64;
    float* bT  = Yt  + NM;
    float* Acc = bT  + NM;
    float* Xt  = Acc + NM;

    const int n4        = NM / 4;                     // 800,000
    const int elemGrid  = (n4 + 255) / 256;           // 3125
    const int gemmGrid  = (JT + 7) / 8;               // 391 blocks x 8 waves
    const int scatGrid  = (EDGES + 7) / 8;            // 50000
    const int trGrid    = (NDIM + 63) / 64;           // 782
    dim3 b256(256);

    // Wp = project_linf(W, kappa)
    project_linf_kernel<<<1, 64, 0, stream>>>(W, Wp);

    // b = (Omega @ U) @ A   (kept transposed as bT [N,64])
    gemm_omega_u_kernel<<<gemmGrid, b256, 0, stream>>>(U, Om, Yt);
    zero_kernel<<<elemGrid, b256, 0, stream>>>((v4f*)bT, n4);
    scatter_kernel<<<scatGrid, b256, 0, stream>>>(Yt, src, dst, val, bT);

    // X0 = b
    copy_kernel<<<elemGrid, b256, 0, stream>>>((const v4f*)bT, (v4f*)Xt, n4);

    // Fixed point: X <- relu((Wp @ X) @ A + b), fw_mitr = 30 iterations
    for (int it = 0; it < ITERS; ++it) {
        gemm_w_x_kernel<<<gemmGrid, b256, 0, stream>>>(Xt, Wp, Yt);
        copy_kernel<<<elemGrid, b256, 0, stream>>>((const v4f*)bT, (v4f*)Acc, n4);
        scatter_kernel<<<scatGrid, b256, 0, stream>>>(Yt, src, dst, val, Acc);
        relu_kernel<<<elemGrid, b256, 0, stream>>>((const v4f*)Acc, (v4f*)Xt, n4);
    }

    // out = Xt^T  ([N,64] -> [64,N])
    transpose_out_kernel<<<trGrid, b256, 0, stream>>>(Xt, out);
}